// _ConvBlock1_43018392436805
// MI455X (gfx1250) — compile-verified
//
#include <hip/hip_runtime.h>

// CDNA5 / gfx1250. fp32 GNN: WMMA f32 16x16x4 for all GEMMs, float4 gathers +
// global_atomic_add_f32 scatters for edge aggregation (bandwidth-bound at
// ~3GB / 23.3TB/s; GEMMs are ~21 GFLOP and negligible on the matrix path).

typedef __attribute__((ext_vector_type(2))) float v2f;
typedef __attribute__((ext_vector_type(8))) float v8f;

#define N_CELLS  200000
#define N_FACES  400000
#define N_POINTS 200000
#define E_CF     800000
#define E_FP     800000
#define E_PP     1200000
#define HID      64
#define OUT_C    128

// ---- WMMA helper: accumulate A[16xK] * B[Kx16] into 16x16 f32 tile --------
// A layout (ISA 7.12.2, 32-bit A 16x4): lane(l16)=M row; v0/v1 = K pair,
// lane-half selects K+0..1 vs K+2..3.  B: lane(l16)=N col, same K striping.
template <int K>
__device__ __forceinline__ v8f wmma_acc(v8f acc,
                                        const float* __restrict__ A, int lda,
                                        const float* __restrict__ B, int ldb,
                                        int l16, int half) {
#pragma unroll
  for (int k0 = 0; k0 < K; k0 += 4) {
    v2f a, b;
    const float* ap = A + l16 * lda + k0 + 2 * half;
    a.x = ap[0];
    a.y = ap[1];
    const float* bp = B + (k0 + 2 * half) * ldb + l16;
    b.x = bp[0];
    b.y = bp[ldb];
    acc = __builtin_amdgcn_wmma_f32_16x16x4_f32(false, a, false, b, (short)0,
                                                acc, false, false);
  }
  return acc;
}

// ---- plain GEMM: D[M,N] = A[M,K] @ B[K,N], one wave per 16x16 tile --------
template <int K>
__global__ void gemm_kernel(const float* __restrict__ A, int lda,
                            const float* __restrict__ B, int ldb,
                            float* __restrict__ D, int ldd,
                            int tilesN, int totalTiles) {
  int wave = (blockIdx.x * blockDim.x + threadIdx.x) >> 5;
  if (wave >= totalTiles) return;  // whole-wave uniform: EXEC stays all-ones
  int lane = threadIdx.x & 31;
  int l16 = lane & 15, half = lane >> 4;
  int tm = wave / tilesN, tn = wave % tilesN;
  v8f acc = {};
  acc = wmma_acc<K>(acc, A + (size_t)tm * 16 * lda, lda, B + tn * 16, ldb,
                    l16, half);
  float* Dp = D + (size_t)tm * 16 * ldd + tn * 16;
#pragma unroll
  for (int r = 0; r < 8; ++r) Dp[(r + 8 * half) * ldd + l16] = acc[r];
}

// ---- concat GEMM: D = [A0 | A1] @ B  (B is (K0+K1) x N) -------------------
template <int K0, int K1>
__global__ void gemm_concat_kernel(const float* __restrict__ A0, int lda0,
                                   const float* __restrict__ A1, int lda1,
                                   const float* __restrict__ B, int ldb,
                                   float* __restrict__ D, int ldd,
                                   int tilesN, int totalTiles) {
  int wave = (blockIdx.x * blockDim.x + threadIdx.x) >> 5;
  if (wave >= totalTiles) return;
  int lane = threadIdx.x & 31;
  int l16 = lane & 15, half = lane >> 4;
  int tm = wave / tilesN, tn = wave % tilesN;
  v8f acc = {};
  acc = wmma_acc<K0>(acc, A0 + (size_t)tm * 16 * lda0, lda0, B + tn * 16, ldb,
                     l16, half);
  acc = wmma_acc<K1>(acc, A1 + (size_t)tm * 16 * lda1, lda1,
                     B + (size_t)K0 * ldb + tn * 16, ldb, l16, half);
  float* Dp = D + (size_t)tm * 16 * ldd + tn * 16;
#pragma unroll
  for (int r = 0; r < 8; ++r) Dp[(r + 8 * half) * ldd + l16] = acc[r];
}

// ---- dual GEMM sharing A: D1 = A@B1, D2 = A@B2 + bias2 --------------------
template <int K>
__global__ void gemm_dual_kernel(const float* __restrict__ A, int lda,
                                 const float* __restrict__ B1,
                                 const float* __restrict__ B2,
                                 const float* __restrict__ bias2, int ldb,
                                 float* __restrict__ D1, float* __restrict__ D2,
                                 int ldd, int tilesN, int totalTiles) {
  int wave = (blockIdx.x * blockDim.x + threadIdx.x) >> 5;
  if (wave >= totalTiles) return;
  int lane = threadIdx.x & 31;
  int l16 = lane & 15, half = lane >> 4;
  int tm = wave / tilesN, tn = wave % tilesN;
  v8f acc1 = {};
  v8f acc2;
  float bv = bias2[tn * 16 + l16];
#pragma unroll
  for (int r = 0; r < 8; ++r) acc2[r] = bv;
  const float* Ap = A + (size_t)tm * 16 * lda;
#pragma unroll
  for (int k0 = 0; k0 < K; k0 += 4) {
    v2f a, b1, b2;
    const float* ap = Ap + l16 * lda + k0 + 2 * half;
    a.x = ap[0];
    a.y = ap[1];
    const float* bp1 = B1 + (k0 + 2 * half) * ldb + tn * 16 + l16;
    b1.x = bp1[0];
    b1.y = bp1[ldb];
    const float* bp2 = B2 + (k0 + 2 * half) * ldb + tn * 16 + l16;
    b2.x = bp2[0];
    b2.y = bp2[ldb];
    acc1 = __builtin_amdgcn_wmma_f32_16x16x4_f32(false, a, false, b1, (short)0,
                                                 acc1, false, false);
    acc2 = __builtin_amdgcn_wmma_f32_16x16x4_f32(false, a, false, b2, (short)0,
                                                 acc2, false, false);
  }
  float* D1p = D1 + (size_t)tm * 16 * ldd + tn * 16;
  float* D2p = D2 + (size_t)tm * 16 * ldd + tn * 16;
#pragma unroll
  for (int r = 0; r < 8; ++r) {
    D1p[(r + 8 * half) * ldd + l16] = acc1[r];
    D2p[(r + 8 * half) * ldd + l16] = acc2[r];
  }
}

// ---- edge scatter: out[dst] += H[src] * ea, CH channels, float4 gather ----
template <int CH>
__global__ void scatter_kernel(const float* __restrict__ H,
                               const int* __restrict__ src,
                               const int* __restrict__ dst,
                               const float* __restrict__ ea,
                               float* __restrict__ out, int nE) {
  constexpr int TPE = CH / 4;
  int gid = blockIdx.x * blockDim.x + threadIdx.x;
  int e = gid / TPE;
  if (e >= nE) return;
  int c4 = (gid % TPE) * 4;
  int s = src[e], d = dst[e];
  float w = ea[e];
  float4 v = *reinterpret_cast<const float4*>(H + (size_t)s * CH + c4);
  float* o = out + (size_t)d * CH + c4;
  atomicAdd(o + 0, v.x * w);
  atomicAdd(o + 1, v.y * w);
  atomicAdd(o + 2, v.z * w);
  atomicAdd(o + 3, v.w * w);
}

// ---- fill rows with a per-channel bias (float4 stores) --------------------
template <int CH>
__global__ void init_bias_kernel(float* __restrict__ out,
                                 const float* __restrict__ bias,
                                 unsigned int nQuads) {
  unsigned int gid = blockIdx.x * blockDim.x + threadIdx.x;
  if (gid >= nQuads) return;
  int c4 = (int)((gid * 4u) % CH);
  float4 b = *reinterpret_cast<const float4*>(bias + c4);
  reinterpret_cast<float4*>(out)[gid] = b;
}

extern "C" void kernel_launch(void* const* d_in, const int* in_sizes, int n_in,
                              void* d_out, int out_size, void* d_ws,
                              size_t ws_size, hipStream_t stream) {
  const float* x_centers = (const float*)d_in[0];
  const float* x_face    = (const float*)d_in[1];
  const float* W_cf      = (const float*)d_in[2];
  const float* b_cf      = (const float*)d_in[3];
  const float* W_fp      = (const float*)d_in[4];
  const float* b_fp      = (const float*)d_in[5];
  const float* W_pp      = (const float*)d_in[6];
  const float* W_pp_root = (const float*)d_in[7];
  const float* b_pp      = (const float*)d_in[8];
  const float* ea_cf     = (const float*)d_in[9];
  const float* ea_fp     = (const float*)d_in[10];
  const float* ea_pp     = (const float*)d_in[11];
  const int* src_cf = (const int*)d_in[12];
  const int* dst_cf = (const int*)d_in[13];
  const int* src_fp = (const int*)d_in[14];
  const int* dst_fp = (const int*)d_in[15];
  const int* src_pp = (const int*)d_in[16];
  const int* dst_pp = (const int*)d_in[17];
  float* out = (float*)d_out;

  float* ws = (float*)d_ws;
  float* h_c    = ws;                                   // [N_CELLS , 64 ]
  float* y_face = h_c + (size_t)N_CELLS * HID;          // [N_FACES , 64 ]
  float* h_f    = y_face + (size_t)N_FACES * HID;       // [N_FACES , 128]
  float* y_pt   = h_f + (size_t)N_FACES * OUT_C;        // [N_POINTS, 128]
  float* t1     = y_pt + (size_t)N_POINTS * OUT_C;      // [N_POINTS, 128]

  const int BT = 256;

  // 1) y_face <- b_cf (bias pre-broadcast for cf scatter)
  init_bias_kernel<HID><<<(N_FACES * HID / 4 + BT - 1) / BT, BT, 0, stream>>>(
      y_face, b_cf, N_FACES * HID / 4);

  // 2) h_c = x_centers @ W_cf        [200k,8]@[8,64]
  {
    int tilesN = HID / 16, tiles = (N_CELLS / 16) * tilesN;
    gemm_kernel<8><<<(tiles * 32 + BT - 1) / BT, BT, 0, stream>>>(
        x_centers, 8, W_cf, HID, h_c, HID, tilesN, tiles);
  }

  // 3) cf scatter: y_face[dst] += h_c[src] * ea_cf
  scatter_kernel<HID><<<(E_CF * (HID / 4) + BT - 1) / BT, BT, 0, stream>>>(
      h_c, src_cf, dst_cf, ea_cf, y_face, E_CF);

  // 4) h_f = [y_face | x_face] @ W_fp   [400k,72]@[72,128]
  {
    int tilesN = OUT_C / 16, tiles = (N_FACES / 16) * tilesN;
    gemm_concat_kernel<HID, 8><<<(tiles * 32 + BT - 1) / BT, BT, 0, stream>>>(
        y_face, HID, x_face, 8, W_fp, OUT_C, h_f, OUT_C, tilesN, tiles);
  }

  // 5) y_pt <- b_fp
  init_bias_kernel<OUT_C><<<(N_POINTS * OUT_C / 4 + BT - 1) / BT, BT, 0,
                            stream>>>(y_pt, b_fp, N_POINTS * OUT_C / 4);

  // 6) fp scatter: y_pt[dst] += h_f[src] * ea_fp
  scatter_kernel<OUT_C><<<(E_FP * (OUT_C / 4) + BT - 1) / BT, BT, 0, stream>>>(
      h_f, src_fp, dst_fp, ea_fp, y_pt, E_FP);

  // 7) t1 = y_pt @ W_pp ; out = y_pt @ W_pp_root + b_pp   (shared A fragment)
  {
    int tilesN = OUT_C / 16, tiles = (N_POINTS / 16) * tilesN;
    gemm_dual_kernel<OUT_C><<<(tiles * 32 + BT - 1) / BT, BT, 0, stream>>>(
        y_pt, OUT_C, W_pp, W_pp_root, b_pp, OUT_C, t1, out, OUT_C, tilesN,
        tiles);
  }

  // 8) pp scatter: out[dst] += t1[src] * ea_pp
  scatter_kernel<OUT_C><<<(E_PP * (OUT_C / 4) + BT - 1) / BT, BT, 0, stream>>>(
      t1, src_pp, dst_pp, ea_pp, out, E_PP);
}